// AdvancedHybridBBBNet_39651138076876
// MI455X (gfx1250) — compile-verified
//
#include <hip/hip_runtime.h>
#include <math.h>

#define NHEADS 8
#define TM 32
#define TN 32
#define KC 16

typedef __attribute__((ext_vector_type(2))) float v2f;
typedef __attribute__((ext_vector_type(8))) float v8f;

// ---------- CDNA5 async global->LDS copy (ASYNCcnt-tracked) ----------
__device__ __forceinline__ void async_b128(void* lds, const float* g) {
  unsigned off = (unsigned)(size_t)lds;   // low 32 bits of generic ptr = LDS offset
  asm volatile("global_load_async_to_lds_b128 %0, %1, off"
               :: "v"(off), "v"(g) : "memory");
}
#define ASYNC_WAIT_2() asm volatile("s_wait_asynccnt 0x2" ::: "memory")
#define ASYNC_WAIT_0() asm volatile("s_wait_asynccnt 0x0" ::: "memory")

// ---------- monotone float<->uint key for atomic max on floats ----------
__device__ __forceinline__ unsigned fkey(float f) {
  int b = __float_as_int(f);
  return (unsigned)(b ^ ((b >> 31) | 0x80000000));
}
__device__ __forceinline__ float funkey(unsigned u) {
  int b = (u & 0x80000000u) ? (int)(u ^ 0x80000000u) : (int)(~u);
  return __int_as_float(b);
}

// ---------- fills ----------
__global__ void fill_f32(float* p, size_t n, float v) {
  size_t i = (size_t)blockIdx.x * blockDim.x + threadIdx.x;
  if (i < n) p[i] = v;
}
__global__ void fill_u32(unsigned* p, size_t n, unsigned v) {
  size_t i = (size_t)blockIdx.x * blockDim.x + threadIdx.x;
  if (i < n) p[i] = v;
}

// ---------- WMMA f32 GEMM, LDS-staged, async double-buffered ----------
// D[M,Nc] = A[M,K] @ B[K,Nc] (+ Cin).  K%16==0, Nc%32==0.
// block = 128 threads (4 waves); 32x32 block tile, 16x16 per wave.
__global__ void gemm_wmma_lds(const float* __restrict__ A,
                              const float* __restrict__ B,
                              const float* __restrict__ Cin,
                              float* __restrict__ D,
                              int M, int K, int Nc) {
  __shared__ float As[2][TM][KC];
  __shared__ float Bs[2][KC][TN];

  const int t    = threadIdx.x;     // 0..127
  const int wave = t >> 5;          // 0..3
  const int lane = t & 31;
  const int half = lane >> 4;       // 0|1
  const int l    = lane & 15;
  const int row0 = blockIdx.y * TM;
  const int col0 = blockIdx.x * TN;
  const int wm   = (wave >> 1) * 16;
  const int wn   = (wave & 1) * 16;
  const bool interior = (row0 + TM <= M);

  // staging assignments: one b128 per thread per matrix chunk
  const int ar = t >> 2;            // 0..31  (A row within tile)
  const int aq = (t & 3) * 4;       // 0,4,8,12 (A col within chunk)
  const int br = t >> 3;            // 0..15  (B row within chunk)
  const int bq = (t & 7) * 4;       // 0..28  (B col within tile)

  const int ocol = col0 + wn + l;   // output column for this lane

  v8f acc;
  if (Cin) {
    for (int v = 0; v < 8; ++v) {
      int r = row0 + wm + v + 8 * half;
      acc[v] = (r < M) ? Cin[(size_t)r * Nc + ocol] : 0.f;
    }
  } else {
    for (int v = 0; v < 8; ++v) acc[v] = 0.f;
  }

  const int nk = K / KC;

  // ---- stage chunk 0 ----
  if (interior) {
    async_b128(&As[0][ar][aq], A + (size_t)(row0 + ar) * K + aq);
    async_b128(&Bs[0][br][bq], B + (size_t)br * Nc + col0 + bq);
  } else {
    int r = row0 + ar;
    float4 va = make_float4(0.f, 0.f, 0.f, 0.f);
    if (r < M) va = *(const float4*)(A + (size_t)r * K + aq);
    *(float4*)&As[0][ar][aq] = va;
    *(float4*)&Bs[0][br][bq] = *(const float4*)(B + (size_t)br * Nc + col0 + bq);
  }

  for (int kc = 0; kc < nk; ++kc) {
    const int cur = kc & 1;
    const int k_next = (kc + 1) * KC;
    if (kc + 1 < nk) {
      if (interior) {
        async_b128(&As[cur ^ 1][ar][aq], A + (size_t)(row0 + ar) * K + k_next + aq);
        async_b128(&Bs[cur ^ 1][br][bq], B + (size_t)(k_next + br) * Nc + col0 + bq);
        ASYNC_WAIT_2();               // current buffer's 2 async ops complete
      } else {
        int r = row0 + ar;
        float4 va = make_float4(0.f, 0.f, 0.f, 0.f);
        if (r < M) va = *(const float4*)(A + (size_t)r * K + k_next + aq);
        *(float4*)&As[cur ^ 1][ar][aq] = va;
        *(float4*)&Bs[cur ^ 1][br][bq] =
            *(const float4*)(B + (size_t)(k_next + br) * Nc + col0 + bq);
      }
    } else {
      if (interior) ASYNC_WAIT_0();
    }
    __syncthreads();

#pragma unroll
    for (int kk = 0; kk < KC; kk += 4) {
      const int ak = kk + 2 * half;
      v2f a, b;
      a.x = As[cur][wm + l][ak];
      a.y = As[cur][wm + l][ak + 1];
      b.x = Bs[cur][ak][wn + l];
      b.y = Bs[cur][ak + 1][wn + l];
      acc = __builtin_amdgcn_wmma_f32_16x16x4_f32(false, a, false, b,
                                                  (short)0, acc, false, false);
    }
    __syncthreads();
  }

  for (int v = 0; v < 8; ++v) {
    int r = row0 + wm + v + 8 * half;
    if (r < M) D[(size_t)r * Nc + ocol] = acc[v];
  }
}

// ---------- GAT1 input transform (K=15, too skinny for WMMA) ----------
__global__ void gat1_transform(const float* __restrict__ x,
                               const float* __restrict__ W,
                               float* __restrict__ h, int C) {
  __shared__ float xs[15];
  int n = blockIdx.x;
  if (threadIdx.x < 15) xs[threadIdx.x] = x[n * 15 + threadIdx.x];
  __syncthreads();
  for (int c = threadIdx.x; c < C; c += blockDim.x) {
    float acc = 0.f;
#pragma unroll
    for (int k = 0; k < 15; ++k) acc += xs[k] * W[k * C + c];
    h[(size_t)n * C + c] = acc;
  }
}

// ---------- per-node attention scores es/ed ----------
__global__ void gat_scores(const float* __restrict__ h,
                           const float* __restrict__ a_s,
                           const float* __restrict__ a_d,
                           float* __restrict__ es, float* __restrict__ ed,
                           int ch) {
  __shared__ float p1[256], p2[256];
  int n = blockIdx.x, t = threadIdx.x;
  int head = t >> 5, lane = t & 31;
  const float* hr = h + (size_t)n * NHEADS * ch + head * ch;
  const float* as_ = a_s + head * ch;
  const float* ad_ = a_d + head * ch;
  float s1 = 0.f, s2 = 0.f;
  for (int c = lane; c < ch; c += 32) { s1 += hr[c] * as_[c]; s2 += hr[c] * ad_[c]; }
  p1[t] = s1; p2[t] = s2;
  __syncthreads();
  if (t < NHEADS) {
    float a = 0.f, b = 0.f;
    for (int i = 0; i < 32; ++i) { a += p1[t * 32 + i]; b += p2[t * 32 + i]; }
    es[n * NHEADS + t] = a;
    ed[n * NHEADS + t] = b;
  }
}

// ---------- edge logits + segment max ----------
__global__ void edge_logits(const int* __restrict__ eidx, int Ee, int Et,
                            const float* __restrict__ es, const float* __restrict__ ed,
                            float* __restrict__ eout, unsigned* __restrict__ emaxu) {
  size_t idx = (size_t)blockIdx.x * blockDim.x + threadIdx.x;
  if (idx >= (size_t)Et * NHEADS) return;
  int e = (int)(idx / NHEADS), hh = (int)(idx % NHEADS);
  int s, d;
  if (e < Ee) { s = eidx[e]; d = eidx[Ee + e]; } else { s = e - Ee; d = s; }
  float v = es[s * NHEADS + hh] + ed[d * NHEADS + hh];
  v = (v > 0.f) ? v : 0.2f * v;            // leaky_relu(0.2)
  eout[idx] = v;
  atomicMax(&emaxu[d * NHEADS + hh], fkey(v));
}

__global__ void decode_emax(const unsigned* __restrict__ u, float* __restrict__ f, size_t n) {
  size_t i = (size_t)blockIdx.x * blockDim.x + threadIdx.x;
  if (i < n) f[i] = (u[i] == 0u) ? 0.f : funkey(u[i]);
}

// ---------- exp(e - max) + segment sum ----------
__global__ void edge_exp_den(const int* __restrict__ eidx, int Ee, int Et,
                             float* __restrict__ ee, const float* __restrict__ emaxf,
                             float* __restrict__ den) {
  size_t idx = (size_t)blockIdx.x * blockDim.x + threadIdx.x;
  if (idx >= (size_t)Et * NHEADS) return;
  int e = (int)(idx / NHEADS), hh = (int)(idx % NHEADS);
  int d = (e < Ee) ? eidx[Ee + e] : (e - Ee);
  float v = __expf(ee[idx] - emaxf[d * NHEADS + hh]);
  ee[idx] = v;
  atomicAdd(&den[d * NHEADS + hh], v);
}

// ---------- GAT scatter: out[dst] += alpha * h[src] ----------
__global__ void gat_scatter(const int* __restrict__ eidx, int Ee,
                            const float* __restrict__ h,
                            const float* __restrict__ ee, const float* __restrict__ den,
                            float* __restrict__ out, int ch) {
  __shared__ float alpha[NHEADS];
  __shared__ int ss, sd;
  int e = blockIdx.x;
  if (threadIdx.x == 0) {
    if (e < Ee) { ss = eidx[e]; sd = eidx[Ee + e]; } else { ss = e - Ee; sd = ss; }
  }
  __syncthreads();
  if (threadIdx.x < NHEADS)
    alpha[threadIdx.x] = ee[(size_t)e * NHEADS + threadIdx.x] /
                         (den[sd * NHEADS + threadIdx.x] + 1e-16f);
  __syncthreads();
  int C = NHEADS * ch;
  const float* hr = h + (size_t)ss * C;
  __builtin_prefetch(hr, 0, 1);                    // -> global_prefetch_b8
  float* orow = out + (size_t)sd * C;
  for (int c = threadIdx.x; c < C; c += blockDim.x)
    atomicAdd(&orow[c], hr[c] * alpha[c / ch]);
}

// ---------- generic scaled scatter (GCN / SAGE aggregation) ----------
__global__ void scatter_edges(const int* __restrict__ eidx, int Ee,
                              const float* __restrict__ h,
                              const float* __restrict__ dinv,
                              float* __restrict__ out, int C) {
  int e = blockIdx.x;
  int s, d;
  if (e < Ee) { s = eidx[e]; d = eidx[Ee + e]; } else { s = e - Ee; d = s; }
  float sc = 1.f;
  if (dinv) sc = dinv[s] * dinv[d];
  const float* hr = h + (size_t)s * C;
  float* orow = out + (size_t)d * C;
  for (int c = threadIdx.x; c < C; c += blockDim.x)
    atomicAdd(&orow[c], hr[c] * sc);
}

__global__ void count_deg(const int* __restrict__ eidx, int Ee, int count, float* deg) {
  int e = blockIdx.x * blockDim.x + threadIdx.x;
  if (e >= count) return;
  int d = (e < Ee) ? eidx[Ee + e] : (e - Ee);
  atomicAdd(&deg[d], 1.f);
}

__global__ void rsqrt_deg(const float* __restrict__ deg, float* __restrict__ dinv, int n) {
  int i = blockIdx.x * blockDim.x + threadIdx.x;
  if (i < n) { float d = deg[i]; dinv[i] = (d > 0.f) ? rsqrtf(d) : 0.f; }
}

__global__ void row_div(float* __restrict__ x, const float* __restrict__ degn, int C) {
  int n = blockIdx.x;
  float inv = 1.f / fmaxf(degn[n], 1.f);
  float* r = x + (size_t)n * C;
  for (int c = threadIdx.x; c < C; c += blockDim.x) r[c] *= inv;
}

// ---------- fused (bias) + LayerNorm + ELU (row-wise) ----------
__global__ void bias_ln_elu(const float* __restrict__ in, const float* __restrict__ bias,
                            const float* __restrict__ g, const float* __restrict__ b,
                            float* __restrict__ out, int C, int do_ln, int do_elu) {
  __shared__ float s1[256], s2[256];
  int row = blockIdx.x, t = threadIdx.x;
  const float* ir = in + (size_t)row * C;
  float sum = 0.f, sq = 0.f;
  for (int c = t; c < C; c += 256) {
    float v = ir[c] + (bias ? bias[c] : 0.f);
    sum += v; sq += v * v;
  }
  s1[t] = sum; s2[t] = sq;
  __syncthreads();
  for (int off = 128; off > 0; off >>= 1) {
    if (t < off) { s1[t] += s1[t + off]; s2[t] += s2[t + off]; }
    __syncthreads();
  }
  float mean = s1[0] / C;
  float var = s2[0] / C - mean * mean;
  float rstd = rsqrtf(var + 1e-5f);
  float* orow = out + (size_t)row * C;
  for (int c = t; c < C; c += 256) {
    float v = ir[c] + (bias ? bias[c] : 0.f);
    float o = do_ln ? ((v - mean) * rstd * g[c] + b[c]) : v;
    if (do_elu) o = (o > 0.f) ? o : expm1f(o);
    orow[c] = o;
  }
}

// ---------- pooling ----------
__global__ void pool_kernel(const float* __restrict__ x, const int* __restrict__ batch,
                            float* __restrict__ xsum, unsigned* __restrict__ xmaxu,
                            float* __restrict__ cnt, int C) {
  int n = blockIdx.x;
  int bg = batch[n];
  if (threadIdx.x == 0) atomicAdd(&cnt[bg], 1.f);
  const float* xr = x + (size_t)n * C;
  for (int c = threadIdx.x; c < C; c += blockDim.x) {
    float v = xr[c];
    atomicAdd(&xsum[(size_t)bg * C + c], v);
    atomicMax(&xmaxu[(size_t)bg * C + c], fkey(v));
  }
}

__global__ void assemble_z(const float* __restrict__ xsum, const unsigned* __restrict__ xmaxu,
                           const float* __restrict__ cnt, float* __restrict__ z, int C) {
  int g = blockIdx.x;
  float inv = 1.f / fmaxf(cnt[g], 1.f);
  for (int c = threadIdx.x; c < C; c += blockDim.x) {
    float s = xsum[(size_t)g * C + c];
    unsigned u = xmaxu[(size_t)g * C + c];
    float m = (u == 0u) ? 0.f : funkey(u);
    z[(size_t)g * 3 * C + c]         = s * inv;  // mean
    z[(size_t)g * 3 * C + C + c]     = m;        // max
    z[(size_t)g * 3 * C + 2 * C + c] = s;        // sum
  }
}

// ---------- final 64->1 head ----------
__global__ void final_head(const float* __restrict__ t4, const float* __restrict__ W,
                           const float* __restrict__ b, float* __restrict__ out, int G) {
  int g = blockIdx.x * blockDim.x + threadIdx.x;
  if (g >= G) return;
  float acc = b[0];
#pragma unroll
  for (int k = 0; k < 64; ++k) acc += t4[(size_t)g * 64 + k] * W[k];
  out[g] = acc;
}

// =====================================================================

static inline dim3 gemm_grid(int M, int Nc) {
  return dim3((Nc + TN - 1) / TN, (M + TM - 1) / TM);
}

extern "C" void kernel_launch(void* const* d_in, const int* in_sizes, int n_in,
                              void* d_out, int out_size, void* d_ws, size_t ws_size,
                              hipStream_t stream) {
  const float* x      = (const float*)d_in[0];
  const int*   eidx   = (const int*)d_in[1];     // [2,E]: src then dst
  const int*   batch  = (const int*)d_in[2];
  const float* gat1_W = (const float*)d_in[3];
  const float* gat1_as= (const float*)d_in[4];
  const float* gat1_ad= (const float*)d_in[5];
  const float* gat1_b = (const float*)d_in[6];
  const float* ln1_g  = (const float*)d_in[7];
  const float* ln1_b  = (const float*)d_in[8];
  const float* gcn_W  = (const float*)d_in[9];
  const float* gcn_b  = (const float*)d_in[10];
  const float* ln2_g  = (const float*)d_in[11];
  const float* ln2_b  = (const float*)d_in[12];
  const float* sage_Wl= (const float*)d_in[13];
  const float* sage_Wr= (const float*)d_in[14];
  const float* sage_b = (const float*)d_in[15];
  const float* ln3_g  = (const float*)d_in[16];
  const float* ln3_b  = (const float*)d_in[17];
  const float* gat2_W = (const float*)d_in[18];
  const float* gat2_as= (const float*)d_in[19];
  const float* gat2_ad= (const float*)d_in[20];
  const float* gat2_b = (const float*)d_in[21];
  const float* ln4_g  = (const float*)d_in[22];
  const float* ln4_b  = (const float*)d_in[23];
  const float* m1_W   = (const float*)d_in[24];
  const float* m1_b   = (const float*)d_in[25];
  const float* m1_g   = (const float*)d_in[26];
  const float* m1_be  = (const float*)d_in[27];
  const float* m2_W   = (const float*)d_in[28];
  const float* m2_b   = (const float*)d_in[29];
  const float* m2_g   = (const float*)d_in[30];
  const float* m2_be  = (const float*)d_in[31];
  const float* m3_W   = (const float*)d_in[32];
  const float* m3_b   = (const float*)d_in[33];
  const float* m3_g   = (const float*)d_in[34];
  const float* m3_be  = (const float*)d_in[35];
  const float* m4_W   = (const float*)d_in[36];
  const float* m4_b   = (const float*)d_in[37];
  const float* m5_W   = (const float*)d_in[38];
  const float* m5_b   = (const float*)d_in[39];
  float* out = (float*)d_out;

  const int Nn = in_sizes[0] / 15;       // 20000
  const int Ee = in_sizes[1] / 2;        // 320000
  const int Et = Ee + Nn;                // with self loops
  const int Gg = out_size;               // 512

  // ---- workspace layout (float elements) ----
  float* W = (float*)d_ws;
  size_t o = 0;
  float* A      = W + o; o += (size_t)Nn * 1024;   // h1, later gcn_h/agg/mean/x3
  float* B      = W + o; o += (size_t)Nn * 1024;   // out1/x1, later h2/out2
  float* es     = W + o; o += (size_t)Nn * NHEADS;
  float* ed     = W + o; o += (size_t)Nn * NHEADS;
  unsigned* emaxu = (unsigned*)(W + o); o += (size_t)Nn * NHEADS;
  float* emaxf  = W + o; o += (size_t)Nn * NHEADS;
  float* den    = W + o; o += (size_t)Nn * NHEADS;
  float* ee     = W + o; o += (size_t)Et * NHEADS;
  float* deg    = W + o; o += Nn;
  float* dinv   = W + o; o += Nn;
  float* degn   = W + o; o += Nn;
  float* cnt    = W + o; o += Gg;
  float* xsum   = W + o; o += (size_t)Gg * 512;
  unsigned* xmaxu = (unsigned*)(W + o); o += (size_t)Gg * 512;
  float* z      = W + o; o += (size_t)Gg * 1536;
  float* t1     = W + o; o += (size_t)Gg * 512;
  float* t2     = W + o; o += (size_t)Gg * 256;
  float* t3     = W + o; o += (size_t)Gg * 128;
  float* t4     = W + o; o += (size_t)Gg * 64;
  (void)ws_size; (void)n_in;

  float* gcn_h    = A;                        // [N,256]
  float* gcn_agg  = A + (size_t)Nn * 256;     // [N,256] -> x2
  float* mean_agg = A + (size_t)Nn * 512;     // [N,256]
  float* x3       = A + (size_t)Nn * 768;     // [N,128]
  float* h2       = B;                        // [N,512]
  float* out2     = B + (size_t)Nn * 512;     // [N,512] -> x4

  auto nb = [](size_t n) { return (unsigned)((n + 255) / 256); };
  const size_t NH = (size_t)Nn * NHEADS;
  const size_t EH = (size_t)Et * NHEADS;

  // ===== GAT1 =====
  gat1_transform<<<Nn, 256, 0, stream>>>(x, gat1_W, A, 1024);
  gat_scores<<<Nn, 256, 0, stream>>>(A, gat1_as, gat1_ad, es, ed, 128);
  fill_u32<<<nb(NH), 256, 0, stream>>>(emaxu, NH, 0u);
  fill_f32<<<nb(NH), 256, 0, stream>>>(den, NH, 0.f);
  fill_f32<<<nb((size_t)Nn * 1024), 256, 0, stream>>>(B, (size_t)Nn * 1024, 0.f);
  edge_logits<<<nb(EH), 256, 0, stream>>>(eidx, Ee, Et, es, ed, ee, emaxu);
  decode_emax<<<nb(NH), 256, 0, stream>>>(emaxu, emaxf, NH);
  edge_exp_den<<<nb(EH), 256, 0, stream>>>(eidx, Ee, Et, ee, emaxf, den);
  gat_scatter<<<Et, 256, 0, stream>>>(eidx, Ee, A, ee, den, B, 128);
  bias_ln_elu<<<Nn, 256, 0, stream>>>(B, gat1_b, ln1_g, ln1_b, B, 1024, 1, 1); // x1

  // ===== GCN =====
  gemm_wmma_lds<<<gemm_grid(Nn, 256), 128, 0, stream>>>(B, gcn_W, nullptr, gcn_h, Nn, 1024, 256);
  fill_f32<<<nb(Nn), 256, 0, stream>>>(deg, Nn, 0.f);
  count_deg<<<nb(Et), 256, 0, stream>>>(eidx, Ee, Et, deg);
  rsqrt_deg<<<nb(Nn), 256, 0, stream>>>(deg, dinv, Nn);
  fill_f32<<<nb((size_t)Nn * 256), 256, 0, stream>>>(gcn_agg, (size_t)Nn * 256, 0.f);
  scatter_edges<<<Et, 256, 0, stream>>>(eidx, Ee, gcn_h, dinv, gcn_agg, 256);
  bias_ln_elu<<<Nn, 256, 0, stream>>>(gcn_agg, gcn_b, ln2_g, ln2_b, gcn_agg, 256, 1, 1); // x2

  // ===== SAGE =====
  fill_f32<<<nb(Nn), 256, 0, stream>>>(degn, Nn, 0.f);
  count_deg<<<nb(Ee), 256, 0, stream>>>(eidx, Ee, Ee, degn);
  fill_f32<<<nb((size_t)Nn * 256), 256, 0, stream>>>(mean_agg, (size_t)Nn * 256, 0.f);
  scatter_edges<<<Ee, 256, 0, stream>>>(eidx, Ee, gcn_agg, nullptr, mean_agg, 256);
  row_div<<<Nn, 256, 0, stream>>>(mean_agg, degn, 256);
  gemm_wmma_lds<<<gemm_grid(Nn, 128), 128, 0, stream>>>(mean_agg, sage_Wl, nullptr, x3, Nn, 256, 128);
  gemm_wmma_lds<<<gemm_grid(Nn, 128), 128, 0, stream>>>(gcn_agg, sage_Wr, x3, x3, Nn, 256, 128);
  bias_ln_elu<<<Nn, 256, 0, stream>>>(x3, sage_b, ln3_g, ln3_b, x3, 128, 1, 1);

  // ===== GAT2 =====
  gemm_wmma_lds<<<gemm_grid(Nn, 512), 128, 0, stream>>>(x3, gat2_W, nullptr, h2, Nn, 128, 512);
  gat_scores<<<Nn, 256, 0, stream>>>(h2, gat2_as, gat2_ad, es, ed, 64);
  fill_u32<<<nb(NH), 256, 0, stream>>>(emaxu, NH, 0u);
  fill_f32<<<nb(NH), 256, 0, stream>>>(den, NH, 0.f);
  fill_f32<<<nb((size_t)Nn * 512), 256, 0, stream>>>(out2, (size_t)Nn * 512, 0.f);
  edge_logits<<<nb(EH), 256, 0, stream>>>(eidx, Ee, Et, es, ed, ee, emaxu);
  decode_emax<<<nb(NH), 256, 0, stream>>>(emaxu, emaxf, NH);
  edge_exp_den<<<nb(EH), 256, 0, stream>>>(eidx, Ee, Et, ee, emaxf, den);
  gat_scatter<<<Et, 256, 0, stream>>>(eidx, Ee, h2, ee, den, out2, 64);
  bias_ln_elu<<<Nn, 256, 0, stream>>>(out2, gat2_b, ln4_g, ln4_b, out2, 512, 1, 1); // x4

  // ===== Pooling =====
  fill_f32<<<nb(Gg), 256, 0, stream>>>(cnt, Gg, 0.f);
  fill_f32<<<nb((size_t)Gg * 512), 256, 0, stream>>>(xsum, (size_t)Gg * 512, 0.f);
  fill_u32<<<nb((size_t)Gg * 512), 256, 0, stream>>>(xmaxu, (size_t)Gg * 512, 0u);
  pool_kernel<<<Nn, 256, 0, stream>>>(out2, batch, xsum, xmaxu, cnt, 512);
  assemble_z<<<Gg, 256, 0, stream>>>(xsum, xmaxu, cnt, z, 512);

  // ===== MLP head =====
  gemm_wmma_lds<<<gemm_grid(Gg, 512), 128, 0, stream>>>(z, m1_W, nullptr, t1, Gg, 1536, 512);
  bias_ln_elu<<<Gg, 256, 0, stream>>>(t1, m1_b, m1_g, m1_be, t1, 512, 1, 1);
  gemm_wmma_lds<<<gemm_grid(Gg, 256), 128, 0, stream>>>(t1, m2_W, nullptr, t2, Gg, 512, 256);
  bias_ln_elu<<<Gg, 256, 0, stream>>>(t2, m2_b, m2_g, m2_be, t2, 256, 1, 1);
  gemm_wmma_lds<<<gemm_grid(Gg, 128), 128, 0, stream>>>(t2, m3_W, nullptr, t3, Gg, 256, 128);
  bias_ln_elu<<<Gg, 256, 0, stream>>>(t3, m3_b, m3_g, m3_be, t3, 128, 1, 1);
  gemm_wmma_lds<<<gemm_grid(Gg, 64), 128, 0, stream>>>(t3, m4_W, nullptr, t4, Gg, 128, 64);
  bias_ln_elu<<<Gg, 256, 0, stream>>>(t4, m4_b, nullptr, nullptr, t4, 64, 0, 1);
  final_head<<<nb(Gg), 256, 0, stream>>>(t4, m5_W, m5_b, out, Gg);
}